// WasserteinMatchingTrainer_59742995087658
// MI455X (gfx1250) — compile-verified
//
#include <hip/hip_runtime.h>
#include <hip/hip_bf16.h>
#include <math.h>

// ---------------------------------------------------------------------------
// Problem constants (match reference): B=4096 points, D=64 features.
// ---------------------------------------------------------------------------
#define BN 4096
#define DK 64
#define NBINS 4096
#define NTOT 16777216ull // 4096*4096

typedef __attribute__((ext_vector_type(2))) float v2f;
typedef __attribute__((ext_vector_type(8))) float v8f;

__device__ __forceinline__ void online_lse(float x, float& m, float& s) {
  if (x > m) { s = s * __expf(m - x) + 1.0f; m = x; }
  else       { s += __expf(x - m); }
}

// ---------------------------------------------------------------------------
// 0) init: zero u, v, hist, loss; min=+MAX_FLT bits, max=0
// ---------------------------------------------------------------------------
__global__ void wm_init(float* u, float* v, unsigned* hist, unsigned* minmax,
                        float* out) {
  int t = blockIdx.x * blockDim.x + threadIdx.x;
  if (t < BN) { u[t] = 0.0f; v[t] = 0.0f; hist[t] = 0u; }
  if (t == 0) { minmax[0] = 0x7f7fffffu; minmax[1] = 0u; out[0] = 0.0f; }
}

// ---------------------------------------------------------------------------
// 1) x_pred = x_source @ W^T + b  via V_WMMA_F32_16X16X4_F32
//    one wave per 16x16 tile; grid = (BN/16, DK/16)
//    A lane L: row i0+(L&15), K = 2*(L>>4)+{0,1}
//    B lane L: col j0+(L&15) -> B[k][n]=W[n][k] -> load W row (j0+L&15)
// ---------------------------------------------------------------------------
__global__ void wm_predict_gemm(const float* __restrict__ xs,
                                const float* __restrict__ W,
                                const float* __restrict__ bias,
                                float* __restrict__ xp) {
  const int i0 = blockIdx.x * 16;
  const int j0 = blockIdx.y * 16;
  const int lane = threadIdx.x;
  const int half = lane >> 4;
  const int l    = lane & 15;
  const float* arow = xs + (size_t)(i0 + l) * DK;
  const float* brow = W  + (size_t)(j0 + l) * DK;
  v8f acc = {};
#pragma unroll
  for (int k0 = 0; k0 < DK; k0 += 4) {
    const int kk = k0 + 2 * half;
    v2f a = *(const v2f*)(arow + kk);
    v2f b = *(const v2f*)(brow + kk);
    acc = __builtin_amdgcn_wmma_f32_16x16x4_f32(false, a, false, b,
                                                (short)0, acc, false, false);
  }
  const float bv = bias[j0 + l];
#pragma unroll
  for (int r = 0; r < 8; ++r) {
    const int row = i0 + r + 8 * half;
    xp[(size_t)row * DK + j0 + l] = acc[r] + bv;
  }
}

// ---------------------------------------------------------------------------
// 2) row squared norms of x_pred and x_target
// ---------------------------------------------------------------------------
__global__ void wm_norms(const float* __restrict__ xp,
                         const float* __restrict__ xt,
                         float* __restrict__ np, float* __restrict__ nt) {
  int r = blockIdx.x * blockDim.x + threadIdx.x;
  if (r < BN) {
    const float* p = xp + (size_t)r * DK;
    float s = 0.0f;
#pragma unroll 8
    for (int k = 0; k < DK; ++k) s = fmaf(p[k], p[k], s);
    np[r] = s;
  } else if (r < 2 * BN) {
    const float* p = xt + (size_t)(r - BN) * DK;
    float s = 0.0f;
#pragma unroll 8
    for (int k = 0; k < DK; ++k) s = fmaf(p[k], p[k], s);
    nt[r - BN] = s;
  }
}

// ---------------------------------------------------------------------------
// 3) cost GEMM: G = xp @ xt^T (f32 WMMA), epilogue:
//    cost = sqrt(max(np+nt-2G, 1e-12)); fused wave min/max -> uint atomics
//    grid = (BN/16, BN/16), one wave per tile
// ---------------------------------------------------------------------------
__global__ void wm_cost_gemm(const float* __restrict__ xp,
                             const float* __restrict__ xt,
                             const float* __restrict__ np,
                             const float* __restrict__ nt,
                             float* __restrict__ cost,
                             unsigned* __restrict__ minmax) {
  const int i0 = blockIdx.x * 16;
  const int j0 = blockIdx.y * 16;
  const int lane = threadIdx.x;
  const int half = lane >> 4;
  const int l    = lane & 15;
  const float* arow = xp + (size_t)(i0 + l) * DK;
  const float* brow = xt + (size_t)(j0 + l) * DK;
  v8f acc = {};
#pragma unroll
  for (int k0 = 0; k0 < DK; k0 += 4) {
    const int kk = k0 + 2 * half;
    v2f a = *(const v2f*)(arow + kk);
    v2f b = *(const v2f*)(brow + kk);
    acc = __builtin_amdgcn_wmma_f32_16x16x4_f32(false, a, false, b,
                                                (short)0, acc, false, false);
  }
  const int col = j0 + l;
  const float ntc = nt[col];
  float mn = 3.4e38f, mx = 0.0f;
#pragma unroll
  for (int r = 0; r < 8; ++r) {
    const int row = i0 + r + 8 * half;
    float sq = np[row] + ntc - 2.0f * acc[r];
    float c  = sqrtf(fmaxf(sq, 1e-12f));
    cost[(size_t)row * BN + col] = c;
    mn = fminf(mn, c);
    mx = fmaxf(mx, c);
  }
  // wave32 shuffle reduction for min/max
#pragma unroll
  for (int off = 16; off > 0; off >>= 1) {
    mn = fminf(mn, __shfl_xor(mn, off, 32));
    mx = fmaxf(mx, __shfl_xor(mx, off, 32));
  }
  if (lane == 0) {
    atomicMin(&minmax[0], __float_as_uint(mn)); // cost >= 0: bit order == fp order
    atomicMax(&minmax[1], __float_as_uint(mx));
  }
}

// ---------------------------------------------------------------------------
// 4) histogram of cost in [min, max], NBINS bins (LDS local hist + merge)
// ---------------------------------------------------------------------------
__global__ void wm_hist(const float* __restrict__ cost,
                        const unsigned* __restrict__ minmax,
                        unsigned* __restrict__ hist) {
  __shared__ unsigned lh[NBINS];
  for (int t = threadIdx.x; t < NBINS; t += blockDim.x) lh[t] = 0u;
  __syncthreads();
  const float mn = __uint_as_float(minmax[0]);
  const float mx = __uint_as_float(minmax[1]);
  const float scale = (float)NBINS / fmaxf(mx - mn, 1e-20f);
  const size_t base = (size_t)blockIdx.x * BN;
  for (int t = threadIdx.x; t < BN; t += blockDim.x) {
    float c = cost[base + t];
    int bin = (int)((c - mn) * scale);
    bin = bin < 0 ? 0 : (bin > NBINS - 1 ? NBINS - 1 : bin);
    atomicAdd(&lh[bin], 1u);
  }
  __syncthreads();
  for (int t = threadIdx.x; t < NBINS; t += blockDim.x)
    if (lh[t]) atomicAdd(&hist[t], lh[t]);
}

// ---------------------------------------------------------------------------
// 5) median scan (1 block): reg = 0.05 * median_est; store reg and -1/reg
// ---------------------------------------------------------------------------
__global__ void wm_median(const unsigned* __restrict__ hist,
                          const unsigned* __restrict__ minmax,
                          float* __restrict__ scal) {
  if (threadIdx.x == 0) {
    const float mn = __uint_as_float(minmax[0]);
    const float mx = __uint_as_float(minmax[1]);
    const unsigned long long target = NTOT / 2ull;
    unsigned long long cum = 0;
    int bin = NBINS - 1;
    for (int k = 0; k < NBINS; ++k) {
      cum += hist[k];
      if (cum >= target) { bin = k; break; }
    }
    const float med = mn + (bin + 0.5f) * (mx - mn) / (float)NBINS;
    const float reg = 0.05f * med;
    scal[0] = reg;
    scal[1] = -1.0f / reg;
  }
}

// ---------------------------------------------------------------------------
// 6a) u[i] = loga - LSE_j( -cost[i,j]/reg + v[j] ) ; one block per row,
//     online LSE per thread + LDS (m,s) tree merge. cost rows live in L2.
// ---------------------------------------------------------------------------
__global__ void wm_update_u(const float* __restrict__ cost,
                            const float* __restrict__ v,
                            float* __restrict__ u,
                            const float* __restrict__ scal) {
  __shared__ float sm[256];
  __shared__ float ss[256];
  const float s = scal[1]; // -1/reg
  const int i = blockIdx.x;
  const float* row = cost + (size_t)i * BN;
  float m = -3.4e38f, acc = 0.0f;
  for (int j = threadIdx.x; j < BN; j += 256) {
    float x = fmaf(row[j], s, v[j]);
    online_lse(x, m, acc);
  }
  sm[threadIdx.x] = m; ss[threadIdx.x] = acc;
  __syncthreads();
  for (int w = 128; w > 0; w >>= 1) {
    if ((int)threadIdx.x < w) {
      float m1 = sm[threadIdx.x], s1 = ss[threadIdx.x];
      float m2 = sm[threadIdx.x + w], s2 = ss[threadIdx.x + w];
      float M = fmaxf(m1, m2);
      ss[threadIdx.x] = s1 * __expf(m1 - M) + s2 * __expf(m2 - M);
      sm[threadIdx.x] = M;
    }
    __syncthreads();
  }
  if (threadIdx.x == 0)
    u[i] = -8.317766166719343f - (sm[0] + __logf(ss[0])); // loga = -log(4096)
}

// ---------------------------------------------------------------------------
// 6b) v[j] = logb - LSE_i( -cost[i,j]/reg + u[i] ) ; block = 64 columns x 4
//     row-slices (coalesced 64-float row segments), merge 4 partials in LDS.
// ---------------------------------------------------------------------------
__global__ void wm_update_v(const float* __restrict__ cost,
                            const float* __restrict__ u,
                            float* __restrict__ v,
                            const float* __restrict__ scal) {
  __shared__ float sm[256];
  __shared__ float ss[256];
  const float s = scal[1];
  const int tx = threadIdx.x & 63;
  const int ty = threadIdx.x >> 6; // 0..3
  const int j  = blockIdx.x * 64 + tx;
  float m = -3.4e38f, acc = 0.0f;
  for (int i = ty; i < BN; i += 4) {
    float x = fmaf(cost[(size_t)i * BN + j], s, u[i]);
    online_lse(x, m, acc);
  }
  sm[ty * 64 + tx] = m; ss[ty * 64 + tx] = acc;
  __syncthreads();
  if (ty == 0) {
    float M = m, S = acc;
#pragma unroll
    for (int p = 1; p < 4; ++p) {
      float m2 = sm[p * 64 + tx], s2 = ss[p * 64 + tx];
      float Mn = fmaxf(M, m2);
      S = S * __expf(M - Mn) + s2 * __expf(m2 - Mn);
      M = Mn;
    }
    v[j] = -8.317766166719343f - (M + __logf(S));
  }
}

// ---------------------------------------------------------------------------
// 7) loss = sum( exp(-cost/reg + u_i + v_j) * cost )
// ---------------------------------------------------------------------------
__global__ void wm_loss(const float* __restrict__ cost,
                        const float* __restrict__ u,
                        const float* __restrict__ v,
                        const float* __restrict__ scal,
                        float* __restrict__ out) {
  __shared__ float red[256];
  const float s = scal[1];
  float acc = 0.0f;
  const size_t stride = (size_t)gridDim.x * 256;
  for (size_t idx = (size_t)blockIdx.x * 256 + threadIdx.x; idx < NTOT;
       idx += stride) {
    const int i = (int)(idx >> 12);
    const int j = (int)(idx & 4095);
    const float c = cost[idx];
    acc += __expf(fmaf(c, s, u[i] + v[j])) * c;
  }
  red[threadIdx.x] = acc;
  __syncthreads();
  for (int w = 128; w > 0; w >>= 1) {
    if ((int)threadIdx.x < w) red[threadIdx.x] += red[threadIdx.x + w];
    __syncthreads();
  }
  if (threadIdx.x == 0) atomicAdd(out, red[0]);
}

// ---------------------------------------------------------------------------
// launch
// ---------------------------------------------------------------------------
extern "C" void kernel_launch(void* const* d_in, const int* in_sizes, int n_in,
                              void* d_out, int out_size, void* d_ws,
                              size_t ws_size, hipStream_t stream) {
  const float* xs = (const float*)d_in[0]; // x_source [4096,64]
  const float* xt = (const float*)d_in[1]; // x_target [4096,64]
  const float* W  = (const float*)d_in[2]; // W [64,64]
  const float* b  = (const float*)d_in[3]; // b [64]
  float* out = (float*)d_out;              // scalar loss

  char* ws = (char*)d_ws;
  size_t off = 0;
  float* cost = (float*)(ws + off); off += (size_t)BN * BN * sizeof(float); // 64MB
  float* xp   = (float*)(ws + off); off += (size_t)BN * DK * sizeof(float);
  float* np   = (float*)(ws + off); off += BN * sizeof(float);
  float* nt   = (float*)(ws + off); off += BN * sizeof(float);
  float* u    = (float*)(ws + off); off += BN * sizeof(float);
  float* v    = (float*)(ws + off); off += BN * sizeof(float);
  unsigned* hist   = (unsigned*)(ws + off); off += NBINS * sizeof(unsigned);
  unsigned* minmax = (unsigned*)(ws + off); off += 256;
  float* scal      = (float*)(ws + off);    off += 256;

  // 0) init
  wm_init<<<(BN + 255) / 256, 256, 0, stream>>>(u, v, hist, minmax, out);
  // 1) predict GEMM (f32 WMMA)
  wm_predict_gemm<<<dim3(BN / 16, DK / 16), 32, 0, stream>>>(xs, W, b, xp);
  // 2) norms
  wm_norms<<<(2 * BN + 255) / 256, 256, 0, stream>>>(xp, xt, np, nt);
  // 3) cost GEMM (f32 WMMA) + min/max
  wm_cost_gemm<<<dim3(BN / 16, BN / 16), 32, 0, stream>>>(xp, xt, np, nt, cost,
                                                          minmax);
  // 4) histogram, 5) median -> reg
  wm_hist<<<BN, 256, 0, stream>>>(cost, minmax, hist);
  wm_median<<<1, 32, 0, stream>>>(hist, minmax, scal);
  // 6) 20 Sinkhorn iterations (L2-resident sweeps)
  for (int it = 0; it < 20; ++it) {
    wm_update_u<<<BN, 256, 0, stream>>>(cost, v, u, scal);
    wm_update_v<<<BN / 64, 256, 0, stream>>>(cost, u, v, scal);
  }
  // 7) loss
  wm_loss<<<4096, 256, 0, stream>>>(cost, u, v, scal, out);
}